// AtomTransformerBlock_884763263523
// MI455X (gfx1250) — compile-verified
//
#include <hip/hip_runtime.h>
#include <math.h>

// ---------------------------------------------------------------------------
// AtomTransformerBlock for MI455X (gfx1250, wave32).
// Matrix ops use V_WMMA_F32_16X16X4_F32 (full fp32, matches reference math).
// ---------------------------------------------------------------------------

typedef __attribute__((ext_vector_type(2))) float v2f;
typedef __attribute__((ext_vector_type(8))) float v8f;

#define N_ATOM   2048
#define C_ATOM   128
#define C_PAIR   16
#define N_HEADS  4
#define BLOCK_SZ 128
#define D_HEAD   32
#define LN_EPS   1e-5f

__device__ __forceinline__ float wave_reduce_sum(float v) {
#pragma unroll
    for (int off = 16; off > 0; off >>= 1) v += __shfl_xor(v, off, 32);
    return v;
}
__device__ __forceinline__ float wave_reduce_max(float v) {
#pragma unroll
    for (int off = 16; off > 0; off >>= 1) v = fmaxf(v, __shfl_xor(v, off, 32));
    return v;
}

// One wave32 per row; each lane owns 4 contiguous floats of the 128-wide row.
__global__ void layernorm_kernel(const float* __restrict__ x,
                                 float* __restrict__ out) {
    const int row  = blockIdx.x;
    const int lane = threadIdx.x;
    const float4 xv = ((const float4*)(x + (size_t)row * C_ATOM))[lane];
    float s = xv.x + xv.y + xv.z + xv.w;
    s = wave_reduce_sum(s);
    const float mu = s * (1.0f / C_ATOM);
    const float d0 = xv.x - mu, d1 = xv.y - mu, d2 = xv.z - mu, d3 = xv.w - mu;
    float ss = d0 * d0 + d1 * d1 + d2 * d2 + d3 * d3;
    ss = wave_reduce_sum(ss);
    const float inv = rsqrtf(ss * (1.0f / C_ATOM) + LN_EPS);
    float4 o;
    o.x = d0 * inv; o.y = d1 * inv; o.z = d2 * inv; o.w = d3 * inv;
    ((float4*)(out + (size_t)row * C_ATOM))[lane] = o;
}

// out[m,n] = act( sum_k A[m,k] * W[n,k] + bias[n] ) + residual[m,n]
// ACT: 0 = identity, 1 = SiLU.
// Per wave: one 16x16 D tile; K-loop in steps of 4 with v_wmma_f32_16x16x4_f32.
// A (16x4) lane layout (ISA 7.12.2): lane L holds M = L%16, K = 2*(L/16)+{0,1}.
// B (4x16, = W^T tile) mirrored: lane L holds N = L%16, K = 2*(L/16)+{0,1}.
// C/D (16x16): VGPR r, lanes 0-15 -> (M=r, N=lane); lanes 16-31 -> (M=r+8).
template <int ACT>
__global__ void wmma_gemm_kernel(const float* __restrict__ A,
                                 const float* __restrict__ W,
                                 float* __restrict__ out,
                                 const float* __restrict__ bias,
                                 const float* __restrict__ residual,
                                 int N, int K) {
    const int lane = threadIdx.x & 31;
    const int wave = threadIdx.x >> 5;
    const int m0   = blockIdx.x * 16;
    const int n0   = (blockIdx.y * 8 + wave) * 16;
    const int li   = lane & 15;
    const int k0   = (lane >> 4) << 1;

    const float* arow = A + (size_t)(m0 + li) * K + k0;
    const float* wrow = W + (size_t)(n0 + li) * K + k0;

    v8f acc = {};
    for (int kk = 0; kk < K; kk += 4) {
        v2f a = *(const v2f*)(arow + kk);
        v2f b = *(const v2f*)(wrow + kk);
        acc = __builtin_amdgcn_wmma_f32_16x16x4_f32(
            /*neg_a=*/false, a, /*neg_b=*/false, b,
            /*c_mod=*/(short)0, acc, /*reuse_a=*/false, /*reuse_b=*/false);
    }

    const int   n     = n0 + li;
    const float bn    = bias ? bias[n] : 0.0f;
    const int   mbase = m0 + ((lane >> 4) << 3);
#pragma unroll
    for (int r = 0; r < 8; ++r) {
        const int m = mbase + r;
        float val = acc[r] + bn;
        if (ACT == 1) val = val / (1.0f + __expf(-val));  // SiLU
        if (residual) val += residual[(size_t)m * N + n];
        out[(size_t)m * N + n] = val;
    }
}

// One 128-thread block per query row i.
__global__ void attn_kernel(const float* __restrict__ q,
                            const float* __restrict__ k,
                            const float* __restrict__ v,
                            const float* __restrict__ pair_emb,
                            const int* __restrict__ block_index,
                            const float* __restrict__ Wb,
                            float* __restrict__ attn_out) {
    __shared__ float q_sh[C_ATOM];
    __shared__ float attn_sh[N_HEADS][BLOCK_SZ];
    __shared__ int   idx_sh[BLOCK_SZ];
    __shared__ float wb_sh[N_HEADS];
    __shared__ float v_sh[32][C_ATOM + 4];  // padded to dodge LDS bank conflicts

    const int i   = blockIdx.x;
    const int tid = threadIdx.x;

    if (tid < N_HEADS) wb_sh[tid] = Wb[tid];      // Wb is [H,1] flat
    q_sh[tid] = q[(size_t)i * C_ATOM + tid];
    const int j = block_index[(size_t)i * BLOCK_SZ + tid];
    idx_sh[tid] = j;
    __syncthreads();

    // pair bias: mean over C_PAIR at (i, j) only (never materialize full NxN)
    const float* pe = pair_emb + ((size_t)i * N_ATOM + (size_t)j) * C_PAIR;
    float ps = 0.f;
#pragma unroll
    for (int t = 0; t < C_PAIR; t += 4) {
        float4 p = *(const float4*)(pe + t);
        ps += p.x + p.y + p.z + p.w;
    }
    const float mean = ps * (1.0f / C_PAIR);

    // scores: per-head dot(q_i, k_j) * scale + mean * Wb[h]
    const float scale = 0.17677669529663687f;  // 1/sqrt(32)
    const float* krow = k + (size_t)j * C_ATOM;
#pragma unroll
    for (int h = 0; h < N_HEADS; ++h) {
        float d = 0.f;
#pragma unroll
        for (int c = h * D_HEAD; c < (h + 1) * D_HEAD; c += 4) {
            float4 kv = *(const float4*)(krow + c);
            float4 qv = *(const float4*)(q_sh + c);
            d += kv.x * qv.x + kv.y * qv.y + kv.z * qv.z + kv.w * qv.w;
        }
        attn_sh[h][tid] = d * scale + mean * wb_sh[h];
    }
    __syncthreads();

    // softmax over the 128 block entries: wave w handles head w
    {
        const int w = tid >> 5, lane = tid & 31;
        float e0 = attn_sh[w][lane];
        float e1 = attn_sh[w][lane + 32];
        float e2 = attn_sh[w][lane + 64];
        float e3 = attn_sh[w][lane + 96];
        float mx = fmaxf(fmaxf(e0, e1), fmaxf(e2, e3));
        mx = wave_reduce_max(mx);
        e0 = __expf(e0 - mx); e1 = __expf(e1 - mx);
        e2 = __expf(e2 - mx); e3 = __expf(e3 - mx);
        const float inv = 1.0f / wave_reduce_sum(e0 + e1 + e2 + e3);
        attn_sh[w][lane]      = e0 * inv;
        attn_sh[w][lane + 32] = e1 * inv;
        attn_sh[w][lane + 64] = e2 * inv;
        attn_sh[w][lane + 96] = e3 * inv;
    }
    __syncthreads();

    // attn_out[i, h*32+d] = sum_b attn[h][b] * v[j_b, h*32+d]
    const int h = tid >> 5, d = tid & 31;
    float acc = 0.f;
    for (int chunk = 0; chunk < 4; ++chunk) {
        // stage 32 gathered v rows into LDS (4 threads per row, float4 each)
        const int rr = tid >> 2, qq = tid & 3;
        const int j2 = idx_sh[chunk * 32 + rr];
        const float* vrow = v + (size_t)j2 * C_ATOM + qq * 32;
        float4* dst = (float4*)(&v_sh[rr][qq * 32]);
#pragma unroll
        for (int t = 0; t < 8; ++t) dst[t] = ((const float4*)vrow)[t];
        __syncthreads();
#pragma unroll
        for (int bb = 0; bb < 32; ++bb)
            acc += attn_sh[h][chunk * 32 + bb] * v_sh[bb][h * D_HEAD + d];
        __syncthreads();
    }
    attn_out[(size_t)i * C_ATOM + tid] = acc;
}

extern "C" void kernel_launch(void* const* d_in, const int* in_sizes, int n_in,
                              void* d_out, int out_size, void* d_ws, size_t ws_size,
                              hipStream_t stream) {
    (void)in_sizes; (void)n_in; (void)out_size; (void)ws_size;

    const float* x           = (const float*)d_in[0];
    const float* pair_emb    = (const float*)d_in[1];
    const int*   block_index = (const int*)d_in[2];
    const float* Wq   = (const float*)d_in[3];
    const float* Wk   = (const float*)d_in[4];
    const float* Wv   = (const float*)d_in[5];
    const float* Wb   = (const float*)d_in[6];
    const float* Wout = (const float*)d_in[7];
    const float* W1   = (const float*)d_in[8];
    const float* b1   = (const float*)d_in[9];
    const float* W2   = (const float*)d_in[10];
    const float* b2   = (const float*)d_in[11];
    float* out = (float*)d_out;

    // workspace carve-up (floats); total = 7*2048*128 + 2048*512 = 2,883,584
    float* ws = (float*)d_ws;
    const size_t S = (size_t)N_ATOM * C_ATOM;
    float* h_buf  = ws;          // layernorm(x)
    float* q_buf  = ws + 1 * S;
    float* k_buf  = ws + 2 * S;
    float* v_buf  = ws + 3 * S;
    float* ao_buf = ws + 4 * S;  // attention output
    float* x1_buf = ws + 5 * S;  // x + attn@Wout^T
    float* h2_buf = ws + 6 * S;  // layernorm(x1)
    float* m1_buf = ws + 7 * S;  // silu(h2@W1^T + b1), [2048, 512]

    const dim3 blk(256);                 // 8 wave32s: 8 n-tiles per block
    const dim3 g128(N_ATOM / 16, 1);     // N=128 -> 8 n-tiles, one block-row
    const dim3 g512(N_ATOM / 16, 4);     // N=512 -> 32 n-tiles

    layernorm_kernel<<<N_ATOM, 32, 0, stream>>>(x, h_buf);

    wmma_gemm_kernel<0><<<g128, blk, 0, stream>>>(h_buf, Wq, q_buf, nullptr, nullptr, 128, 128);
    wmma_gemm_kernel<0><<<g128, blk, 0, stream>>>(h_buf, Wk, k_buf, nullptr, nullptr, 128, 128);
    wmma_gemm_kernel<0><<<g128, blk, 0, stream>>>(h_buf, Wv, v_buf, nullptr, nullptr, 128, 128);

    attn_kernel<<<N_ATOM, 128, 0, stream>>>(q_buf, k_buf, v_buf, pair_emb,
                                            block_index, Wb, ao_buf);

    // x1 = x + attn_out @ Wout^T
    wmma_gemm_kernel<0><<<g128, blk, 0, stream>>>(ao_buf, Wout, x1_buf, nullptr, x, 128, 128);

    layernorm_kernel<<<N_ATOM, 32, 0, stream>>>(x1_buf, h2_buf);

    // mlp1 = silu(h2 @ W1^T + b1)
    wmma_gemm_kernel<1><<<g512, blk, 0, stream>>>(h2_buf, W1, m1_buf, b1, nullptr, 512, 128);

    // out = x1 + mlp1 @ W2^T + b2
    wmma_gemm_kernel<0><<<g128, blk, 0, stream>>>(m1_buf, W2, out, b2, x1_buf, 128, 512);
}